// BinaryClassificationLTC_65712999629562
// MI455X (gfx1250) — compile-verified
//
#include <hip/hip_runtime.h>
#include <hip/hip_bf16.h>

typedef __attribute__((ext_vector_type(2))) float v2f;
typedef __attribute__((ext_vector_type(8))) float v8f;

#ifndef __has_builtin
#define __has_builtin(x) 0
#endif
#if defined(__gfx1250__) && __has_builtin(__builtin_amdgcn_wmma_f32_16x16x4_f32)
#define USE_WMMA_F32X4 1
#else
#define USE_WMMA_F32X4 0
#endif

// Packed synapse record: one ds_load_b128 per (i,j) pair.
struct __align__(16) Syn { float sig, mu, w, we; };

// ---------------------------------------------------------------------------
// Kernel 1: LTC recurrence. 256 threads = 4 batches x 64 neurons.
// All parameter matrices resident in LDS (packed float4), v-state in LDS.
// Produces seq[B,T] (motor neuron trace) into workspace.
// ---------------------------------------------------------------------------
__global__ __launch_bounds__(256) void ltc_seq_kernel(
    const float* __restrict__ x,        // [B,T,64]
    const float* __restrict__ input_w,  // [64]
    const float* __restrict__ input_b,  // [64]
    const float* __restrict__ s_mu_g,   // [64,64]
    const float* __restrict__ s_sig_g,
    const float* __restrict__ s_w_g,
    const float* __restrict__ s_erev_g,
    const float* __restrict__ r_mu_g,   // [64,64]
    const float* __restrict__ r_sig_g,
    const float* __restrict__ r_w_g,
    const float* __restrict__ r_erev_g,
    const float* __restrict__ gleak_g,  // [64]
    const float* __restrict__ vleak_g,
    const float* __restrict__ cm_g,
    const float* __restrict__ out_w,    // [1]
    const float* __restrict__ out_b,    // [1]
    const float* __restrict__ s_mask_g, // [64,64]
    const float* __restrict__ r_mask_g, // [64,64]
    float* __restrict__ seq,            // [B,T]
    int T)
{
    extern __shared__ char ldsraw[];
    Syn*   smat = (Syn*)ldsraw;            // 64*64 sensory records (64KB)
    Syn*   rmat = smat + 64 * 64;          // 64*64 recurrent records (64KB)
    float* vsh  = (float*)(rmat + 64 * 64); // 4*64 state
    float* xt   = vsh + 4 * 64;             // 4*64 mapped input

    const int tid = threadIdx.x;

    // Load + pre-combine parameters into LDS once per block.
    for (int idx = tid; idx < 64 * 64; idx += 256) {
        Syn s;
        s.sig = s_sig_g[idx]; s.mu = s_mu_g[idx];
        float sw = s_w_g[idx] * s_mask_g[idx];
        s.w = sw; s.we = sw * s_erev_g[idx];
        smat[idx] = s;
        Syn r;
        r.sig = r_sig_g[idx]; r.mu = r_mu_g[idx];
        float rw = r_w_g[idx] * r_mask_g[idx];
        r.w = rw; r.we = rw * r_erev_g[idx];
        rmat[idx] = r;
    }

    const int bl = tid >> 6;     // batch slot 0..3
    const int j  = tid & 63;     // neuron (and input-channel slot)
    const int b  = blockIdx.x * 4 + bl;

    vsh[tid] = 0.0f;             // v0 = 0

    const float gl   = gleak_g[j];
    const float glvl = gl * vleak_g[j];
    const float cmt  = cm_g[j] * 6.0f;      // cm * ode_unfolds
    const float iw   = input_w[j];
    const float ib   = input_b[j];
    const float ow   = out_w[0];
    const float ob   = out_b[0];
    __syncthreads();

    const float* xb = x + (size_t)b * T * 64;
    float* vrow = vsh + bl * 64;
    const float* xrow = xt + bl * 64;

    for (int t = 0; t < T; ++t) {
        xt[tid] = xb[t * 64 + j] * iw + ib;
        if (t + 1 < T) __builtin_prefetch(xb + (t + 1) * 64 + j, 0, 1);
        __syncthreads();

        // Sensory contraction over D=64 (constant within the step).
        float num_s = 0.0f, den_s = 0.0f;
        #pragma unroll 4
        for (int d = 0; d < 64; ++d) {
            Syn s = smat[d * 64 + j];
            float z  = s.sig * (xrow[d] - s.mu);
            float sg = __builtin_amdgcn_rcpf(1.0f + __expf(-z));
            num_s = fmaf(s.we, sg, num_s);
            den_s = fmaf(s.w,  sg, den_s);
        }
        const float num_base = glvl + num_s;
        const float den_base = cmt + gl + den_s + 1e-8f;

        // 6 fused-solver unfolds over the recurrent synapses.
        for (int u = 0; u < 6; ++u) {
            float num_r = 0.0f, den_r = 0.0f;
            const float vj = vrow[j];
            #pragma unroll 4
            for (int i = 0; i < 64; ++i) {
                Syn r = rmat[i * 64 + j];
                float z  = r.sig * (vrow[i] - r.mu);
                float sg = __builtin_amdgcn_rcpf(1.0f + __expf(-z));
                num_r = fmaf(r.we, sg, num_r);
                den_r = fmaf(r.w,  sg, den_r);
            }
            const float vnew = (fmaf(cmt, vj, num_base) + num_r) / (den_base + den_r);
            __syncthreads();
            vrow[j] = vnew;
            __syncthreads();
        }
        if (j == 0) seq[(size_t)b * T + t] = fmaf(vrow[0], ow, ob);
    }
}

// ---------------------------------------------------------------------------
// Kernel 2: head GEMM + reduce. One block = 16 batches, 8 waves.
// Each wave computes 16x16 tiles of h = relu(seq @ fc1^T + b1) via
// V_WMMA_F32_16X16X4_F32 (K-loop over T), then block-reduces with fc2.
// A frag: lane l holds M=l&15, K = k0 + 2*(l>>4) + {0,1}. B symmetric in N.
// ---------------------------------------------------------------------------
__global__ __launch_bounds__(256) void ltc_fc_kernel(
    const float* __restrict__ seq,   // [B,T]
    const float* __restrict__ fc1_w, // [H,T]
    const float* __restrict__ fc1_b, // [H]
    const float* __restrict__ fc2_w, // [H]
    const float* __restrict__ fc2_b, // [1]
    float* __restrict__ out,         // [B]
    int T, int H)
{
    extern __shared__ float lds[];
    float* hbuf    = lds;            // 16*H
    float* partial = lds + 16 * H;   // 256
    const int tid   = threadIdx.x;
    const int mbase = blockIdx.x * 16;

#if USE_WMMA_F32X4
    const int wave = tid >> 5, lane = tid & 31;
    const int half = lane >> 4, l16 = lane & 15;
    const float* Arow = seq + (size_t)(mbase + l16) * T;
    for (int nb = wave * 16; nb < H; nb += 128) {
        const int n = nb + l16;
        const float* Brow = fc1_w + (size_t)n * T;   // B[k][n] = fc1_w[n][k]
        v8f c = {};
        for (int k0 = 0; k0 < T; k0 += 4) {
            const int k = k0 + 2 * half;
            v2f a; a.x = Arow[k]; a.y = Arow[k + 1];
            v2f bfrag; bfrag.x = Brow[k]; bfrag.y = Brow[k + 1];
            c = __builtin_amdgcn_wmma_f32_16x16x4_f32(
                    false, a, false, bfrag, (short)0, c, false, false);
        }
        const float bn = fc1_b[n];
        #pragma unroll
        for (int r = 0; r < 8; ++r) {
            const int m = r + 8 * half;               // D layout: vgpr r, half
            hbuf[m * H + n] = fmaxf(c[r] + bn, 0.0f);
        }
    }
#else
    // Scalar fallback (compiles everywhere; wmma=0 signals missing builtin).
    for (int idx = tid; idx < 16 * H; idx += 256) {
        const int m = idx / H, n = idx % H;
        const float* Arow = seq + (size_t)(mbase + m) * T;
        const float* Brow = fc1_w + (size_t)n * T;
        float acc = fc1_b[n];
        for (int k = 0; k < T; ++k) acc = fmaf(Arow[k], Brow[k], acc);
        hbuf[m * H + n] = fmaxf(acc, 0.0f);
    }
#endif
    __syncthreads();
    {
        const int row = tid >> 4, seg = tid & 15;
        const int chunk = (H + 15) >> 4;
        const int n0 = seg * chunk;
        const int n1 = (n0 + chunk < H) ? (n0 + chunk) : H;
        float s = 0.0f;
        for (int n = n0; n < n1; ++n) s = fmaf(hbuf[row * H + n], fc2_w[n], s);
        partial[tid] = s;
    }
    __syncthreads();
    if (tid < 16) {
        float s = fc2_b[0];
        #pragma unroll
        for (int q = 0; q < 16; ++q) s += partial[tid * 16 + q];
        out[mbase + tid] = 1.0f / (1.0f + __expf(-s));
    }
}

// ---------------------------------------------------------------------------
extern "C" void kernel_launch(void* const* d_in, const int* in_sizes, int n_in,
                              void* d_out, int out_size, void* d_ws, size_t ws_size,
                              hipStream_t stream)
{
    const float* x       = (const float*)d_in[0];
    // d_in[1] = states (unused by reference forward)
    const float* input_w = (const float*)d_in[2];
    const float* input_b = (const float*)d_in[3];
    const float* s_mu    = (const float*)d_in[4];
    const float* s_sig   = (const float*)d_in[5];
    const float* s_w     = (const float*)d_in[6];
    const float* s_erev  = (const float*)d_in[7];
    const float* r_mu    = (const float*)d_in[8];
    const float* r_sig   = (const float*)d_in[9];
    const float* r_w     = (const float*)d_in[10];
    const float* r_erev  = (const float*)d_in[11];
    const float* gleak   = (const float*)d_in[12];
    const float* vleak   = (const float*)d_in[13];
    const float* cm      = (const float*)d_in[14];
    const float* out_w   = (const float*)d_in[15];
    const float* out_b   = (const float*)d_in[16];
    const float* fc1_w   = (const float*)d_in[17];
    const float* fc1_b   = (const float*)d_in[18];
    const float* fc2_w   = (const float*)d_in[19];
    const float* fc2_b   = (const float*)d_in[20];
    const float* s_mask  = (const float*)d_in[21];
    const float* r_mask  = (const float*)d_in[22];

    const int Dn = in_sizes[2];            // 64
    const int Hn = in_sizes[18];           // 128
    const int Tn = in_sizes[17] / Hn;      // 256
    const int Bn = in_sizes[0] / (Tn * Dn);// 1024

    float* seq = (float*)d_ws;             // [B,T] scratch, fully overwritten

    const size_t lds1 = (size_t)(2 * 64 * 64) * sizeof(Syn)
                      + (size_t)(2 * 4 * 64) * sizeof(float);   // ~130 KB
    ltc_seq_kernel<<<Bn / 4, 256, lds1, stream>>>(
        x, input_w, input_b, s_mu, s_sig, s_w, s_erev,
        r_mu, r_sig, r_w, r_erev, gleak, vleak, cm,
        out_w, out_b, s_mask, r_mask, seq, Tn);

    const size_t lds2 = (size_t)(16 * Hn + 256) * sizeof(float);
    ltc_fc_kernel<<<Bn / 16, 256, lds2, stream>>>(
        seq, fc1_w, fc1_b, fc2_w, fc2_b, (float*)d_out, Tn, Hn);
}